// StSkillGNN_9259949490765
// MI455X (gfx1250) — compile-verified
//
#include <hip/hip_runtime.h>
#include <hip/hip_bf16.h>

typedef float v2f __attribute__((ext_vector_type(2)));
typedef float v8f __attribute__((ext_vector_type(8)));

#define DIM   128
#define TSEQ  12
#define PRESERVE 0.1f

// ---------------------------------------------------------------------------
// 1) deg[n] = 1.0 (self-loop weight), cnt[n] = 0 (CSR edge counts)
__global__ void k_init(float* __restrict__ deg, int* __restrict__ cnt, int n) {
    int i = blockIdx.x * blockDim.x + threadIdx.x;
    if (i < n) { deg[i] = 1.0f; cnt[i] = 0; }
}

// 2) weighted degree + integer in-degree per destination
__global__ void k_deg_cnt(const int* __restrict__ dst, const float* __restrict__ w,
                          float* __restrict__ deg, int* __restrict__ cnt, int e) {
    int i = blockIdx.x * blockDim.x + threadIdx.x;
    if (i < e) {
        int d = dst[i];
        atomicAdd(&deg[d], w[i]);
        atomicAdd(&cnt[d], 1);
    }
}

// 3) dinv = deg>0 ? rsqrt(deg) : 0  (in place), selfnorm = dinv^2
__global__ void k_dinv(float* __restrict__ deg_dinv, float* __restrict__ selfnorm, int n) {
    int i = blockIdx.x * blockDim.x + threadIdx.x;
    if (i < n) {
        float d = deg_dinv[i];
        float r = (d > 0.0f) ? rsqrtf(d) : 0.0f;
        deg_dinv[i] = r;
        selfnorm[i] = r * r;
    }
}

// 4) single-block exclusive prefix sum: rowptr[0..n] from cnt; cursor = rowptr copy
__global__ __launch_bounds__(1024)
void k_scan(const int* __restrict__ cnt, int* __restrict__ rowptr,
            int* __restrict__ cursor, int n) {
    __shared__ int sh[1024];
    __shared__ int s_base;
    int tid = threadIdx.x;
    if (tid == 0) s_base = 0;
    __syncthreads();
    for (int base = 0; base < n; base += 1024) {
        int i = base + tid;
        int v = (i < n) ? cnt[i] : 0;
        sh[tid] = v;
        __syncthreads();
        for (int off = 1; off < 1024; off <<= 1) {
            int add = (tid >= off) ? sh[tid - off] : 0;
            __syncthreads();
            sh[tid] += add;
            __syncthreads();
        }
        int incl = sh[tid];
        int b = s_base;
        if (i < n) { rowptr[i] = b + incl - v; cursor[i] = b + incl - v; }
        __syncthreads();
        if (tid == 1023) s_base = b + incl;   // running total
        __syncthreads();
    }
    if (tid == 0) rowptr[n] = s_base;
}

// 5) fill CSR: edge -> slot in its destination's row; store src index and
//    precomputed gcn_norm = dinv[src]*w*dinv[dst]
__global__ void k_fill(const int* __restrict__ src, const int* __restrict__ dst,
                       const float* __restrict__ w, const float* __restrict__ dinv,
                       int* __restrict__ cursor, int* __restrict__ csr_src,
                       float* __restrict__ csr_nrm, int e) {
    int i = blockIdx.x * blockDim.x + threadIdx.x;
    if (i < e) {
        int s = src[i], d = dst[i];
        int pos = atomicAdd(&cursor[d], 1);
        csr_src[pos] = s;
        csr_nrm[pos] = dinv[s] * w[i] * dinv[d];
    }
}

// ---------------------------------------------------------------------------
// WMMA f32 GEMM:  C[n,d] = sum_k A(n,k) * W[d,k]  (+bias[d] if bias!=null)
// MODE 0: A = a0, row-major [N, KTOT]
// MODE 1: A(n,k) = k<128 ? emb[n,k] : k<256 ? demand[n,T-1,k-128] : supply[n,T-1,k-256]
// Block: 256 threads = 8 waves; 16 rows x 128 cols per block; wave w owns the
// 16x16 tile at columns [16w, 16w+16). V_WMMA_F32_16X16X4_F32, EXEC all-ones.
template <int KTOT, int MODE>
__global__ __launch_bounds__(256)
void k_wmma_gemm(const float* __restrict__ a0, const float* __restrict__ a1,
                 const float* __restrict__ a2, const float* __restrict__ W,
                 const float* __restrict__ bias, float* __restrict__ C) {
    __shared__ float As[16][36];    // 16 rows x 32 K (pad 36: 16B-aligned rows, no bank conflicts)
    __shared__ float Ws[128][36];   // 128 d  x 32 K

    const int row0 = blockIdx.x * 16;
    const int tid  = threadIdx.x;
    const int wave = tid >> 5;
    const int lane = tid & 31;
    const int half = lane >> 4;
    const int l    = lane & 15;
    const int d0   = wave * 16;

    v8f acc = {0.f, 0.f, 0.f, 0.f, 0.f, 0.f, 0.f, 0.f};

    for (int k0 = 0; k0 < KTOT; k0 += 32) {
        // ---- stage A slab (float4, chunk-uniform source select for MODE 1)
        const float* ap;
        int astride;
        if constexpr (MODE == 0) {
            ap = a0 + k0;                                   astride = KTOT;
        } else {
            if (k0 < DIM)          { ap = a0 + k0;                              astride = DIM; }
            else if (k0 < 2 * DIM) { ap = a1 + (TSEQ - 1) * DIM + (k0 - DIM);   astride = TSEQ * DIM; }
            else                   { ap = a2 + (TSEQ - 1) * DIM + (k0 - 2*DIM); astride = TSEQ * DIM; }
        }
        for (int idx = tid; idx < 16 * 8; idx += 256) {
            int r = idx >> 3, kq = (idx & 7) << 2;
            float4 v = *(const float4*)&ap[(size_t)(row0 + r) * astride + kq];
            *(float4*)&As[r][kq] = v;
        }
        // ---- stage W slab (float4)
        for (int idx = tid; idx < 128 * 8; idx += 256) {
            int d = idx >> 3, kq = (idx & 7) << 2;
            *(float4*)&Ws[d][kq] = *(const float4*)&W[(size_t)d * KTOT + k0 + kq];
        }
        __syncthreads();

        // ISA 16x4 f32 A layout: lanes 0-15 carry {K=kk,kk+1}, lanes 16-31 {kk+2,kk+3};
        // B (4x16) mirrors it with N across lanes; B[k,n] = W[d0+n, k].
        #pragma unroll
        for (int kk = 0; kk < 32; kk += 4) {
            v2f a, b;
            a.x = As[l][kk + 2 * half + 0];
            a.y = As[l][kk + 2 * half + 1];
            b.x = Ws[d0 + l][kk + 2 * half + 0];
            b.y = Ws[d0 + l][kk + 2 * half + 1];
            acc = __builtin_amdgcn_wmma_f32_16x16x4_f32(
                false, a, false, b, (short)0, acc, false, false);
        }
        __syncthreads();
    }

    // C/D layout: vgpr v, lanes 0-15 -> (M=v, N=l); lanes 16-31 -> (M=v+8, N=l)
    float bv = (bias != nullptr) ? bias[d0 + l] : 0.0f;
    #pragma unroll
    for (int v = 0; v < 8; ++v) {
        int m = v + 8 * half;
        C[(size_t)(row0 + m) * DIM + d0 + l] = acc[v] + bv;
    }
}

// ---------------------------------------------------------------------------
// 6) fused per-layer aggregation: one wave per destination node.
//    nxt[n] = 0.9*( sum_{e in row(n)} nrm_e * x[src_e]  + selfnorm[n]*x[n] + b )
//           + 0.1*temp[n]
//    Pure gather (no atomics); x is 10MB -> L2-resident on the 192MB L2.
__global__ __launch_bounds__(256)
void k_aggregate(const int* __restrict__ rowptr, const int* __restrict__ csr_src,
                 const float* __restrict__ csr_nrm, const float* __restrict__ x,
                 const float* __restrict__ selfnorm, const float* __restrict__ bias,
                 const float* __restrict__ temp, float* __restrict__ nxt, int n) {
    int node = blockIdx.x * (blockDim.x >> 5) + (threadIdx.x >> 5);
    if (node >= n) return;                       // wave-uniform exit
    int lane = threadIdx.x & 31;

    const float4* x4 = (const float4*)x;
    int beg = rowptr[node];
    int end = rowptr[node + 1];

    float4 acc = {0.f, 0.f, 0.f, 0.f};
    int j = beg;
    for (; j + 1 < end; j += 2) {                // unroll-2 for load latency
        int   s0 = csr_src[j],     s1 = csr_src[j + 1];
        float w0 = csr_nrm[j],     w1 = csr_nrm[j + 1];
        float4 v0 = x4[(size_t)s0 * 32 + lane];
        float4 v1 = x4[(size_t)s1 * 32 + lane];
        acc.x += w0 * v0.x + w1 * v1.x;
        acc.y += w0 * v0.y + w1 * v1.y;
        acc.z += w0 * v0.z + w1 * v1.z;
        acc.w += w0 * v0.w + w1 * v1.w;
    }
    if (j < end) {
        int   s0 = csr_src[j];
        float w0 = csr_nrm[j];
        float4 v0 = x4[(size_t)s0 * 32 + lane];
        acc.x += w0 * v0.x; acc.y += w0 * v0.y;
        acc.z += w0 * v0.z; acc.w += w0 * v0.w;
    }

    float  sn = selfnorm[node];
    float4 xv = x4[(size_t)node * 32 + lane];
    float4 bv = ((const float4*)bias)[lane];
    float4 tv = ((const float4*)temp)[(size_t)node * 32 + lane];
    float4 o;
    o.x = (1.0f - PRESERVE) * (acc.x + sn * xv.x + bv.x) + PRESERVE * tv.x;
    o.y = (1.0f - PRESERVE) * (acc.y + sn * xv.y + bv.y) + PRESERVE * tv.y;
    o.z = (1.0f - PRESERVE) * (acc.z + sn * xv.z + bv.z) + PRESERVE * tv.z;
    o.w = (1.0f - PRESERVE) * (acc.w + sn * xv.w + bv.w) + PRESERVE * tv.w;
    ((float4*)nxt)[(size_t)node * 32 + lane] = o;
}

// ---------------------------------------------------------------------------
extern "C" void kernel_launch(void* const* d_in, const int* in_sizes, int n_in,
                              void* d_out, int out_size, void* d_ws, size_t ws_size,
                              hipStream_t stream) {
    const float* demand  = (const float*)d_in[0];
    const float* supply  = (const float*)d_in[1];
    const int*   eidx    = (const int*)  d_in[5];
    const float* eattr   = (const float*)d_in[6];
    const float* emb     = (const float*)d_in[10];
    const float* fuse_W  = (const float*)d_in[11];
    const float* fuse_b  = (const float*)d_in[12];
    const float* gcn_W   = (const float*)d_in[13];
    const float* gcn_b   = (const float*)d_in[14];

    const int N = in_sizes[10] / DIM;
    const int E = in_sizes[5] / 2;
    const int L = in_sizes[13] / (DIM * DIM);
    const int* src = eidx;          // edge_index[0]
    const int* dst = eidx + E;      // edge_index[1]

    // carve workspace (256B aligned slices)
    char* wsp = (char*)d_ws;
    auto carve = [&](size_t bytes) -> void* {
        char* p = wsp;
        wsp += (bytes + 255) & ~(size_t)255;
        return p;
    };
    float* dinv     = (float*)carve((size_t)N * 4);        // deg -> dinv in place
    float* selfnorm = (float*)carve((size_t)N * 4);
    int*   cnt      = (int*)  carve((size_t)N * 4);
    int*   rowptr   = (int*)  carve((size_t)(N + 1) * 4);
    int*   cursor   = (int*)  carve((size_t)N * 4);
    int*   csr_src  = (int*)  carve((size_t)E * 4);
    float* csr_nrm  = (float*)carve((size_t)E * 4);
    float* tA       = (float*)carve((size_t)N * DIM * 4);  // temp ping
    float* tB       = (float*)carve((size_t)N * DIM * 4);  // temp pong
    float* xb       = (float*)carve((size_t)N * DIM * 4);  // x = temp @ W^T

    const int TPB = 256;
    // ---- graph prep (once, reused by all layers)
    k_init   <<<(N + TPB - 1) / TPB, TPB, 0, stream>>>(dinv, cnt, N);
    k_deg_cnt<<<(E + TPB - 1) / TPB, TPB, 0, stream>>>(dst, eattr, dinv, cnt, E);
    k_dinv   <<<(N + TPB - 1) / TPB, TPB, 0, stream>>>(dinv, selfnorm, N);
    k_scan   <<<1, 1024, 0, stream>>>(cnt, rowptr, cursor, N);
    k_fill   <<<(E + TPB - 1) / TPB, TPB, 0, stream>>>(src, dst, eattr, dinv,
                                                       cursor, csr_src, csr_nrm, E);

    // ---- fuse GEMM: [N,384] @ [384,128] + bias -> tA
    k_wmma_gemm<3 * DIM, 1><<<N / 16, 256, 0, stream>>>(emb, demand, supply, fuse_W, fuse_b, tA);

    // ---- L GCN layers: WMMA GEMM + atomic-free CSR aggregation (fused residual)
    float* cur = tA;
    for (int i = 0; i < L; ++i) {
        k_wmma_gemm<DIM, 0><<<N / 16, 256, 0, stream>>>(
            cur, nullptr, nullptr, gcn_W + (size_t)i * DIM * DIM, nullptr, xb);
        float* nxt = (i == L - 1) ? (float*)d_out : (cur == tA ? tB : tA);
        k_aggregate<<<(N + 7) / 8, 256, 0, stream>>>(
            rowptr, csr_src, csr_nrm, xb, selfnorm, gcn_b + (size_t)i * DIM, cur, nxt, N);
        cur = nxt;
    }
}